// ModelWrapper_44890998178119
// MI455X (gfx1250) — compile-verified
//
#include <hip/hip_runtime.h>
#include <hip/hip_bf16.h>

typedef __attribute__((ext_vector_type(2))) float v2f;
typedef __attribute__((ext_vector_type(8))) float v8f;

#define BATCH 8192
#define QDIM 29
#define LDIM 5
#define TPTS 101
#define CW 32
#define OW 64
#define NSUB 4
#define ROWS_PER_WAVE 16
#define WAVES_PER_BLOCK 4
#define ODE_BLOCKS (BATCH / (ROWS_PER_WAVE * WAVES_PER_BLOCK)) // 128
#define WSTRIDE 68   // padded column stride (floats) to avoid 64-bank conflicts
#define HSTRIDE 68

// ---- Tsit5 tableau (Tsitouras 2011), matches reference ----
__device__ const float ATAB[6][5] = {
  {0.f, 0.f, 0.f, 0.f, 0.f},
  {0.161f, 0.f, 0.f, 0.f, 0.f},
  {-0.008480655492356989f, 0.335480655492357f, 0.f, 0.f, 0.f},
  {2.8971530571054935f, -6.359448489975075f, 4.3622954328695815f, 0.f, 0.f},
  {5.325864828439257f, -11.748883564062828f, 7.4955393428898365f, -0.09249506636175525f, 0.f},
  {5.86145544294642f, -12.92096931784711f, 8.159367898576159f, -0.071584973281401f, -0.028269050394068383f}
};
__device__ const float BTAB[6] = {
  0.09646076681806523f, 0.01f, 0.4798896504144996f,
  1.379008574103742f, -3.290069515436081f, 2.324710524099774f
};

__device__ __forceinline__ float tanh_f(float x) {
  float a = fabsf(x);
  float e = __expf(-2.0f * a);
  float t = (1.0f - e) / (1.0f + e);
  return x < 0.0f ? -t : t;
}

// One ODE function evaluation for 16 batch rows owned by this wave.
// Yin/Fout: per-wave LDS [16*5]. H: per-wave LDS staging [16*HSTRIDE].
// sW0: 5x64 row-major; sWmid: 4 layers 64x64 column-major (stride WSTRIDE);
// sW5: 5x64 (out-major); sBias: b0(64) b1..b4(64 each) b5(5).
__device__ __forceinline__ void ode_f(int lane, const float* Yin, float* Fout,
                                      float* H, const float* sW0,
                                      const float* sWmid, const float* sW5,
                                      const float* sBias,
                                      float reg, float inv_reg)
{
  // ---- layer 0: 5 -> 64 (VALU) ----
  for (int i = lane; i < ROWS_PER_WAVE * OW; i += 32) {
    int m = i >> 6, c = i & 63;
    float a = sBias[c];
#pragma unroll
    for (int k = 0; k < LDIM; ++k) a = fmaf(Yin[m * LDIM + k], sW0[k * OW + c], a);
    H[m * HSTRIDE + c] = fmaxf(a, 0.0f);
  }
  __builtin_amdgcn_wave_barrier();

  const int arow = lane & 15;
  const int asub = (lane & 16) ? 2 : 0;   // low lanes: K0/K1, high lanes: K2/K3
  const int mbase = (lane & 16) ? 8 : 0;  // C/D layout row base

  // ---- layers 1..4: 64 -> 64 via V_WMMA_F32_16X16X4_F32 ----
  for (int l = 0; l < 4; ++l) {
    const float* Wl = sWmid + l * (64 * WSTRIDE);
    const float* bl = sBias + 64 + l * 64;

    v2f A[16];  // full 16x64 A operand for this wave
#pragma unroll
    for (int ks = 0; ks < 16; ++ks)
      A[ks] = *(const v2f*)&H[arow * HSTRIDE + ks * 4 + asub];
    __builtin_amdgcn_wave_barrier();

#pragma unroll
    for (int t = 0; t < 4; ++t) {
      int col = t * 16 + (lane & 15);
      float bv = bl[col];
      v8f acc;
#pragma unroll
      for (int r = 0; r < 8; ++r) acc[r] = bv;
#pragma unroll
      for (int ks = 0; ks < 16; ++ks) {
        v2f Bf = *(const v2f*)&Wl[col * WSTRIDE + ks * 4 + asub];
        acc = __builtin_amdgcn_wmma_f32_16x16x4_f32(
            false, A[ks], false, Bf, (short)0, acc, false, false);
      }
#pragma unroll
      for (int r = 0; r < 8; ++r)
        H[(mbase + r) * HSTRIDE + col] = fmaxf(acc[r], 0.0f);
    }
    __builtin_amdgcn_wave_barrier();
  }

  // ---- layer 5: 64 -> 5, f = reg * tanh(out / reg) ----
  for (int i = lane; i < ROWS_PER_WAVE * LDIM; i += 32) {
    int m = i / LDIM, c = i % LDIM;
    float a = sBias[5 * 64 + c];
#pragma unroll 8
    for (int k = 0; k < OW; ++k) a = fmaf(H[m * HSTRIDE + k], sW5[c * OW + k], a);
    Fout[i] = reg * tanh_f(a * inv_reg);
  }
  __builtin_amdgcn_wave_barrier();
}

__global__ __launch_bounds__(WAVES_PER_BLOCK * 32)
void ode_kernel(const float* __restrict__ t_range,
                const float* __restrict__ W0, const float* __restrict__ b0,
                const float* __restrict__ W1, const float* __restrict__ b1,
                const float* __restrict__ W2, const float* __restrict__ b2,
                const float* __restrict__ W3, const float* __restrict__ b3,
                const float* __restrict__ W4, const float* __restrict__ b4,
                const float* __restrict__ W5, const float* __restrict__ b5,
                const float* __restrict__ regp,
                float* traj /* [B][T][5]; slice t=0 holds z0 on entry */)
{
  __shared__ __align__(16) float sWmid[4 * 64 * WSTRIDE];
  __shared__ __align__(16) float sW0[LDIM * OW];
  __shared__ __align__(16) float sW5[LDIM * OW];
  __shared__ __align__(16) float sBias[328];
  __shared__ __align__(16) float sH[WAVES_PER_BLOCK][ROWS_PER_WAVE * HSTRIDE];
  __shared__ __align__(16) float sK[WAVES_PER_BLOCK][6 * 80];
  __shared__ __align__(16) float sY[WAVES_PER_BLOCK][80];
  __shared__ __align__(16) float sYT[WAVES_PER_BLOCK][80];

  const int tid = threadIdx.x;

  // ---- cooperative weight staging ----
  const float* Wmids[4] = {W1, W2, W3, W4};
  for (int l = 0; l < 4; ++l) {
    const float* W = Wmids[l];
    for (int i = tid; i < 4096; i += blockDim.x) {
      int n = i >> 6, k = i & 63;
      sWmid[l * (64 * WSTRIDE) + n * WSTRIDE + k] = W[k * 64 + n]; // column-major
    }
  }
  for (int i = tid; i < LDIM * OW; i += blockDim.x) sW0[i] = W0[i];
  for (int i = tid; i < LDIM * OW; i += blockDim.x) {
    int c = i >> 6, k = i & 63;
    sW5[i] = W5[k * LDIM + c]; // out-major
  }
  if (tid < 64) {
    sBias[tid]       = b0[tid];
    sBias[64 + tid]  = b1[tid];
    sBias[128 + tid] = b2[tid];
    sBias[192 + tid] = b3[tid];
    sBias[256 + tid] = b4[tid];
  }
  if (tid < 5) sBias[320 + tid] = b5[tid];
  __syncthreads();

  const float reg = regp[0];
  const float inv_reg = 1.0f / reg;

  const int wave = tid >> 5, lane = tid & 31;
  const int row0 = (blockIdx.x * WAVES_PER_BLOCK + wave) * ROWS_PER_WAVE;
  float* H  = sH[wave];
  float* K  = sK[wave];
  float* Y  = sY[wave];
  float* YT = sYT[wave];

  // init y from traj[:, 0, :]
  for (int i = lane; i < 80; i += 32) {
    int m = i / 5, c = i % 5;
    Y[i] = traj[(size_t)(row0 + m) * (TPTS * LDIM) + c];
  }
  __builtin_amdgcn_wave_barrier();

  for (int it = 0; it < TPTS - 1; ++it) {
    float h = (t_range[it + 1] - t_range[it]) * (1.0f / NSUB);
    for (int s = 0; s < NSUB; ++s) {
      for (int st = 0; st < 6; ++st) {
        const float* in = Y;
        if (st > 0) {
          for (int i = lane; i < 80; i += 32) {
            float acc = 0.0f;
            for (int j = 0; j < st; ++j) acc = fmaf(ATAB[st][j], K[j * 80 + i], acc);
            YT[i] = fmaf(h, acc, Y[i]);
          }
          __builtin_amdgcn_wave_barrier();
          in = YT;
        }
        ode_f(lane, in, K + st * 80, H, sW0, sWmid, sW5, sBias, reg, inv_reg);
      }
      for (int i = lane; i < 80; i += 32) {
        float acc = 0.0f;
#pragma unroll
        for (int j = 0; j < 6; ++j) acc = fmaf(BTAB[j], K[j * 80 + i], acc);
        Y[i] = fmaf(h, acc, Y[i]);
      }
      __builtin_amdgcn_wave_barrier();
    }
    for (int i = lane; i < 80; i += 32) {
      int m = i / 5, c = i % 5;
      traj[(size_t)(row0 + m) * (TPTS * LDIM) + (it + 1) * LDIM + c] = Y[i];
    }
  }
}

// ---- encoder: x0[B,29] -> traj[:,0,:5] (tanh) ----
__global__ void enc_kernel(const float* __restrict__ x0,
                           const float* __restrict__ W0, const float* __restrict__ b0,
                           const float* __restrict__ W1, const float* __restrict__ b1,
                           const float* __restrict__ W2, const float* __restrict__ b2,
                           const float* __restrict__ W3, const float* __restrict__ b3,
                           float* __restrict__ traj)
{
  int b = blockIdx.x * blockDim.x + threadIdx.x;
  if (b >= BATCH) return;
  const float* x = x0 + (size_t)b * QDIM;
  float xr[QDIM];
#pragma unroll
  for (int k = 0; k < QDIM; ++k) xr[k] = x[k];
  float h1[CW], h2[CW];
#pragma unroll
  for (int c = 0; c < CW; ++c) {
    float a = b0[c];
#pragma unroll
    for (int k = 0; k < QDIM; ++k) a = fmaf(xr[k], W0[k * CW + c], a);
    h1[c] = fmaxf(a, 0.0f);
  }
#pragma unroll
  for (int c = 0; c < CW; ++c) {
    float a = b1[c];
#pragma unroll
    for (int k = 0; k < CW; ++k) a = fmaf(h1[k], W1[k * CW + c], a);
    h2[c] = fmaxf(a, 0.0f);
  }
#pragma unroll
  for (int c = 0; c < CW; ++c) {
    float a = b2[c];
#pragma unroll
    for (int k = 0; k < CW; ++k) a = fmaf(h2[k], W2[k * CW + c], a);
    h1[c] = fmaxf(a, 0.0f);  // reuse h1 as h3
  }
#pragma unroll
  for (int c = 0; c < LDIM; ++c) {
    float a = b3[c];
#pragma unroll
    for (int k = 0; k < CW; ++k) a = fmaf(h1[k], W3[k * LDIM + c], a);
    traj[(size_t)b * (TPTS * LDIM) + c] = tanh_f(a);
  }
}

// ---- decoder: traj[B*T,5] -> out[B*T,29] (tanh) ----
__global__ void dec_kernel(const float* __restrict__ traj,
                           const float* __restrict__ W0, const float* __restrict__ b0,
                           const float* __restrict__ W1, const float* __restrict__ b1,
                           const float* __restrict__ W2, const float* __restrict__ b2,
                           const float* __restrict__ W3, const float* __restrict__ b3,
                           float* __restrict__ out)
{
  int r = blockIdx.x * blockDim.x + threadIdx.x;
  if (r >= BATCH * TPTS) return;
  const float* in = traj + (size_t)r * LDIM;
  float xr[LDIM];
#pragma unroll
  for (int k = 0; k < LDIM; ++k) xr[k] = in[k];
  float h1[CW], h2[CW];
#pragma unroll
  for (int c = 0; c < CW; ++c) {
    float a = b0[c];
#pragma unroll
    for (int k = 0; k < LDIM; ++k) a = fmaf(xr[k], W0[k * CW + c], a);
    h1[c] = fmaxf(a, 0.0f);
  }
#pragma unroll
  for (int c = 0; c < CW; ++c) {
    float a = b1[c];
#pragma unroll
    for (int k = 0; k < CW; ++k) a = fmaf(h1[k], W1[k * CW + c], a);
    h2[c] = fmaxf(a, 0.0f);
  }
#pragma unroll
  for (int c = 0; c < CW; ++c) {
    float a = b2[c];
#pragma unroll
    for (int k = 0; k < CW; ++k) a = fmaf(h2[k], W2[k * CW + c], a);
    h1[c] = fmaxf(a, 0.0f);  // reuse h1 as h3
  }
#pragma unroll
  for (int c = 0; c < QDIM; ++c) {
    float a = b3[c];
#pragma unroll
    for (int k = 0; k < CW; ++k) a = fmaf(h1[k], W3[k * QDIM + c], a);
    out[(size_t)r * QDIM + c] = tanh_f(a);
  }
}

extern "C" void kernel_launch(void* const* d_in, const int* in_sizes, int n_in,
                              void* d_out, int out_size, void* d_ws, size_t ws_size,
                              hipStream_t stream)
{
  (void)in_sizes; (void)n_in; (void)out_size; (void)ws_size;
  const float* x0      = (const float*)d_in[0];
  const float* t_range = (const float*)d_in[1];
  const float* eW[4]; const float* eB[4];
  for (int i = 0; i < 4; ++i) { eW[i] = (const float*)d_in[2 + 2 * i];  eB[i] = (const float*)d_in[3 + 2 * i]; }
  const float* oW[6]; const float* oB[6];
  for (int i = 0; i < 6; ++i) { oW[i] = (const float*)d_in[10 + 2 * i]; oB[i] = (const float*)d_in[11 + 2 * i]; }
  const float* dW[4]; const float* dB[4];
  for (int i = 0; i < 4; ++i) { dW[i] = (const float*)d_in[22 + 2 * i]; dB[i] = (const float*)d_in[23 + 2 * i]; }
  const float* reg = (const float*)d_in[30];

  float* traj = (float*)d_ws; // BATCH * TPTS * LDIM floats (~16.5 MB)

  enc_kernel<<<BATCH / 256, 256, 0, stream>>>(
      x0, eW[0], eB[0], eW[1], eB[1], eW[2], eB[2], eW[3], eB[3], traj);

  ode_kernel<<<ODE_BLOCKS, WAVES_PER_BLOCK * 32, 0, stream>>>(
      t_range,
      oW[0], oB[0], oW[1], oB[1], oW[2], oB[2],
      oW[3], oB[3], oW[4], oB[4], oW[5], oB[5],
      reg, traj);

  int rows = BATCH * TPTS;
  dec_kernel<<<(rows + 255) / 256, 256, 0, stream>>>(
      traj, dW[0], dB[0], dW[1], dB[1], dW[2], dB[2], dW[3], dB[3],
      (float*)d_out);
}